// RoPE_Gen_15144054686381
// MI455X (gfx1250) — compile-verified
//
#include <hip/hip_runtime.h>

typedef __attribute__((ext_vector_type(2))) float v2f;
typedef __attribute__((ext_vector_type(4))) float f4;
typedef __attribute__((ext_vector_type(8))) float v8f;

#define HEAD_DIM 128
#define HALF     64
#define CACHE_LEN 4096
#define FULL_LEN  4097
#define BLK      128
#define LSTRIDE  132   // padded LDS row stride (keeps 16B alignment, decouples banks)

__global__ __launch_bounds__(128)
void flash_decode_rope(const float* __restrict__ q,
                       const float* __restrict__ k,
                       const float* __restrict__ v,
                       const float* __restrict__ cK,
                       const float* __restrict__ cV,
                       const float* __restrict__ fc,
                       float* __restrict__ res,
                       float* __restrict__ kf,
                       float* __restrict__ vf)
{
    __shared__ float Kt[HEAD_DIM * LSTRIDE];  // [d][j] tile of cache_K
    __shared__ float Vt[BLK * LSTRIDE];       // [j][d] tile of cache_V
    __shared__ float qs[HEAD_DIM];            // rotated q
    __shared__ float p[BLK];                  // block scores / probs
    __shared__ float red[BLK];                // reduction scratch

    const int t    = threadIdx.x;
    const int bh   = blockIdx.x;          // b*32 + h
    const int lane = t & 31;
    const int wave = t >> 5;
    const int half = lane >> 4;           // lane 0-15 vs 16-31
    const int m    = lane & 15;           // row within 16x16 tile

    const size_t qoff  = (size_t)bh * HEAD_DIM;
    const size_t ckoff = (size_t)bh * HEAD_DIM * CACHE_LEN;
    const size_t cvoff = (size_t)bh * CACHE_LEN * HEAD_DIM;
    const size_t kfoff = (size_t)bh * HEAD_DIM * FULL_LEN;
    const size_t vfoff = (size_t)bh * FULL_LEN * HEAD_DIM;

    // ---------------- position 0: RoPE(q,k), write k_full/v_full pos 0 ----------------
    float v0 = v[qoff + t];
    vf[vfoff + t] = v0;              // v_full[b,h,0,t]
    float accd = v0;                 // online-softmax accumulator for dim d = t (weight exp(0)=1)

    if (t < HALF) {
        float f  = fc[t];
        float qh = q[qoff + t] * f;
        float kh = k[qoff + t] * f;
        qs[t] = qh;  qs[t + HALF] = qh;                       // duplicated halves
        kf[kfoff + (size_t)t         * FULL_LEN] = kh;        // k_full[b,h,t,0]
        kf[kfoff + (size_t)(t + HALF) * FULL_LEN] = kh;       // k_full[b,h,t+64,0]
        red[t] = 2.0f * qh * kh;                              // both halves contribute
    } else {
        red[t] = 0.0f;
    }
    __syncthreads();
    for (int s = 64; s > 0; s >>= 1) { if (t < s) red[t] += red[t + s]; __syncthreads(); }
    float m_run = red[0] * (1.0f / 128.0f);   // score of position 0 (div by head_dim, not sqrt)
    float s_run = 1.0f;
    __syncthreads();

    const float inv_hd = 1.0f / 128.0f;

    // ---------------- stream 32 blocks of 128 cache positions ----------------
    for (int j0 = 0; j0 < CACHE_LEN; j0 += BLK) {
        // K tile: global -> LDS, write-through to k_full (rows odd-offset -> scalar stores)
        #pragma unroll 4
        for (int it = 0; it < 32; ++it) {
            int idx = t + 128 * it;          // float4 index within 128x128 tile
            int d   = idx >> 5;              // 32 float4 per row
            int jj  = (idx & 31) << 2;
            f4 kv = *(const f4*)(cK + ckoff + (size_t)d * CACHE_LEN + j0 + jj);
            *(f4*)(&Kt[d * LSTRIDE + jj]) = kv;
            size_t ko = kfoff + (size_t)d * FULL_LEN + 1 + j0 + jj;
            kf[ko + 0] = kv.x; kf[ko + 1] = kv.y; kf[ko + 2] = kv.z; kf[ko + 3] = kv.w;
        }
        // V tile: global -> LDS, write-through to v_full (16B aligned)
        #pragma unroll 4
        for (int it = 0; it < 32; ++it) {
            int idx = t + 128 * it;
            int j   = idx >> 5;
            int dd  = (idx & 31) << 2;
            f4 vv = *(const f4*)(cV + cvoff + (size_t)(j0 + j) * HEAD_DIM + dd);
            *(f4*)(&Vt[j * LSTRIDE + dd]) = vv;
            *(f4*)(vf + vfoff + (size_t)(1 + j0 + j) * HEAD_DIM + dd) = vv;
        }
        __syncthreads();

        // Scores via V_WMMA_F32_16X16X4_F32: D = (K^T tile 16x4) x (q chunk 4x16) + C
        // B columns all equal -> D independent of N; accumulate over 32 chunks of the d-axis.
        #pragma unroll
        for (int tt = 0; tt < 2; ++tt) {
            int jt = (wave * 2 + tt) * 16;   // this wave's 16 positions
            v8f c = {0.f, 0.f, 0.f, 0.f, 0.f, 0.f, 0.f, 0.f};
            for (int kk = 0; kk < 32; ++kk) {
                int d0 = kk * 4 + 2 * half;  // A/B layout: K-index = vgpr + 2*lane_half
                v2f a, b;
                a.x = Kt[(d0    ) * LSTRIDE + jt + m];
                a.y = Kt[(d0 + 1) * LSTRIDE + jt + m];
                b.x = qs[d0];
                b.y = qs[d0 + 1];
                c = __builtin_amdgcn_wmma_f32_16x16x4_f32(
                        false, a, false, b, (short)0, c, false, false);
            }
            if (m == 0) {                    // lanes 0 & 16 hold M = r + 8*half, any N
                #pragma unroll
                for (int r = 0; r < 8; ++r) p[jt + r + 8 * half] = c[r] * inv_hd;
            }
        }
        __syncthreads();

        // Online softmax update over this block
        float x = p[t];
        red[t] = x; __syncthreads();
        for (int s = 64; s > 0; s >>= 1) { if (t < s) red[t] = fmaxf(red[t], red[t + s]); __syncthreads(); }
        float bm    = red[0];
        float m_new = fmaxf(m_run, bm);
        float pe    = __expf(x - m_new);
        p[t] = pe;
        __syncthreads();
        red[t] = pe; __syncthreads();
        for (int s = 64; s > 0; s >>= 1) { if (t < s) red[t] += red[t + s]; __syncthreads(); }
        float s_blk = red[0];
        float scale = __expf(m_run - m_new);
        s_run = s_run * scale + s_blk;
        accd *= scale;
        for (int j = 0; j < BLK; ++j)        // thread t owns output dim d=t; conflict-free LDS reads
            accd += p[j] * Vt[j * LSTRIDE + t];
        m_run = m_new;
        __syncthreads();                     // protect Kt/Vt/p before next block's loads
    }

    res[qoff + t] = accd / s_run;            // res[b,0,h,t]
}

extern "C" void kernel_launch(void* const* d_in, const int* in_sizes, int n_in,
                              void* d_out, int out_size, void* d_ws, size_t ws_size,
                              hipStream_t stream) {
    const float* q  = (const float*)d_in[0];
    const float* k  = (const float*)d_in[1];
    const float* v  = (const float*)d_in[2];
    const float* cK = (const float*)d_in[3];
    const float* cV = (const float*)d_in[4];
    const float* fc = (const float*)d_in[5];

    float* out = (float*)d_out;
    float* res = out;                                        // [8,1,32,128]
    float* kf  = out + (size_t)32768;                        // [8,32,128,4097]
    float* vf  = out + (size_t)32768 + (size_t)256 * 128 * 4097;  // [8,32,4097,128]

    flash_decode_rope<<<dim3(256), dim3(128), 0, stream>>>(q, k, v, cK, cV, fc, res, kf, vf);
}